// PyGAggregationProcessor_71880572665978
// MI455X (gfx1250) — compile-verified
//
#include <hip/hip_runtime.h>

#define HIDDEN   128
#define LN_EPS_F 1e-5f

typedef float v2f __attribute__((ext_vector_type(2)));
typedef float v8f __attribute__((ext_vector_type(8)));

// ---------- order-preserving float<->uint mapping (for hardware u32 atomic max) ----------
__device__ __forceinline__ unsigned enc_f32(float f) {
  unsigned b = __float_as_uint(f);
  return (b & 0x80000000u) ? ~b : (b | 0x80000000u);
}
__device__ __forceinline__ float dec_f32(unsigned u) {
  unsigned b = (u & 0x80000000u) ? (u & 0x7fffffffu) : ~u;
  return __uint_as_float(b);
}

// ---------------------------------------------------------------------------
// Kernel 1: hx = H @ W^T + bias   (exact fp32 via V_WMMA_F32_16X16X4_F32)
// One wave -> 16 rows x 128 cols. A-tile loaded once, reused for 8 N-tiles.
// ---------------------------------------------------------------------------
__global__ void __launch_bounds__(256) gemm_bias_wmma(
    const float* __restrict__ H, const float* __restrict__ Wm,
    const float* __restrict__ bias, float* __restrict__ out, int n_rows)
{
  const int wave = (int)((blockIdx.x * blockDim.x + threadIdx.x) >> 5);
  const int lane = (int)(threadIdx.x & 31);
  const int row0 = wave * 16;
  if (row0 >= n_rows) return;                 // wave-uniform: EXEC stays all-ones

  const int m  = lane & 15;                   // row within tile (A) / col within tile (B)
  const int kk = (lane >> 4) << 1;            // K sub-offset: lanes 0-15 -> K+0/1, 16-31 -> K+2/3
  const int mh = (lane >> 4) << 3;            // C/D: upper half lanes hold M = r+8

  const float* hrow = H + (size_t)(row0 + m) * HIDDEN;
  v2f a[32];
#pragma unroll
  for (int ks = 0; ks < 32; ++ks)             // A 16x128 slab, K step 4
    a[ks] = *(const v2f*)(hrow + ks * 4 + kk);

#pragma unroll
  for (int nt = 0; nt < 8; ++nt) {
    const int col = nt * 16 + m;
    const float* wrow = Wm + (size_t)col * HIDDEN;   // B[k][n] = W[n][k] (contiguous in k)
    v8f c = (v8f){0.f, 0.f, 0.f, 0.f, 0.f, 0.f, 0.f, 0.f};
#pragma unroll
    for (int ks = 0; ks < 32; ++ks) {
      v2f bvec = *(const v2f*)(wrow + ks * 4 + kk);
      c = __builtin_amdgcn_wmma_f32_16x16x4_f32(
              /*neg_a=*/false, a[ks], /*neg_b=*/false, bvec,
              /*c_mod=*/(short)0, c, /*reuse_a=*/false, /*reuse_b=*/false);
    }
    const float bv = bias[col];
#pragma unroll
    for (int r = 0; r < 8; ++r)
      out[(size_t)(row0 + r + mh) * HIDDEN + col] = c[r] + bv;
  }
}

// ---------------------------------------------------------------------------
// Kernel 2: agg = enc(hx)   (self-loop contribution; also initializes max)
// ---------------------------------------------------------------------------
__global__ void __launch_bounds__(256) init_agg(
    const float* __restrict__ hx, unsigned* __restrict__ agg, int n_vec4)
{
  int i = blockIdx.x * blockDim.x + threadIdx.x;
  if (i >= n_vec4) return;
  float4 v = ((const float4*)hx)[i];
  uint4 e;
  e.x = enc_f32(v.x); e.y = enc_f32(v.y); e.z = enc_f32(v.z); e.w = enc_f32(v.w);
  ((uint4*)agg)[i] = e;
}

// ---------------------------------------------------------------------------
// Kernel 3: scatter-max over edges. One wave per edge (float4/lane),
// grid-stride with prefetch of the next gather row (global_prefetch_b8).
// ---------------------------------------------------------------------------
__global__ void __launch_bounds__(256) scatter_max_edges(
    const float* __restrict__ hx, const int* __restrict__ src,
    const int* __restrict__ dst, unsigned* __restrict__ agg, int n_edges)
{
  const int lane    = (int)(threadIdx.x & 31);
  const int wave    = (int)((blockIdx.x * blockDim.x + threadIdx.x) >> 5);
  const int n_waves = (int)((gridDim.x * blockDim.x) >> 5);

  for (int e = wave; e < n_edges; e += n_waves) {
    const int en = e + n_waves;
    if (en < n_edges)                          // hide gather latency of next trip
      __builtin_prefetch(hx + (size_t)src[en] * HIDDEN + lane * 4, 0, 0);

    const int s = src[e];
    const int d = dst[e];
    float4 v = ((const float4*)(hx + (size_t)s * HIDDEN))[lane];
    unsigned* q = agg + (size_t)d * HIDDEN + lane * 4;
    atomicMax(q + 0, enc_f32(v.x));
    atomicMax(q + 1, enc_f32(v.y));
    atomicMax(q + 2, enc_f32(v.z));
    atomicMax(q + 3, enc_f32(v.w));
  }
}

// ---------------------------------------------------------------------------
// Kernel 4: h = LayerNorm(dec(agg) + residual) * gamma + beta  [+ ReLU]
// One wave per node row; wave32 butterfly reduction for mean/var.
// ---------------------------------------------------------------------------
__global__ void __launch_bounds__(256) ln_residual(
    const unsigned* __restrict__ agg, const float* __restrict__ res,
    const float* __restrict__ gamma, const float* __restrict__ beta,
    float* __restrict__ out, int n_nodes, int do_relu)
{
  const int node = (int)((blockIdx.x * blockDim.x + threadIdx.x) >> 5);
  const int lane = (int)(threadIdx.x & 31);
  if (node >= n_nodes) return;

  uint4  e = ((const uint4*)(agg + (size_t)node * HIDDEN))[lane];
  float4 r = ((const float4*)(res + (size_t)node * HIDDEN))[lane];
  float x0 = dec_f32(e.x) + r.x;
  float x1 = dec_f32(e.y) + r.y;
  float x2 = dec_f32(e.z) + r.z;
  float x3 = dec_f32(e.w) + r.w;

  float s  = x0 + x1 + x2 + x3;
  float s2 = x0 * x0 + x1 * x1 + x2 * x2 + x3 * x3;
#pragma unroll
  for (int off = 16; off > 0; off >>= 1) {
    s  += __shfl_xor(s,  off, 32);
    s2 += __shfl_xor(s2, off, 32);
  }
  const float mu   = s  * (1.f / HIDDEN);
  const float var  = s2 * (1.f / HIDDEN) - mu * mu;
  const float rstd = rsqrtf(var + LN_EPS_F);

  float4 g = ((const float4*)gamma)[lane];
  float4 b = ((const float4*)beta)[lane];
  float y0 = (x0 - mu) * rstd * g.x + b.x;
  float y1 = (x1 - mu) * rstd * g.y + b.y;
  float y2 = (x2 - mu) * rstd * g.z + b.z;
  float y3 = (x3 - mu) * rstd * g.w + b.w;
  if (do_relu) {
    y0 = fmaxf(y0, 0.f); y1 = fmaxf(y1, 0.f);
    y2 = fmaxf(y2, 0.f); y3 = fmaxf(y3, 0.f);
  }
  ((float4*)(out + (size_t)node * HIDDEN))[lane] = make_float4(y0, y1, y2, y3);
}

// ---------------------------------------------------------------------------
extern "C" void kernel_launch(void* const* d_in, const int* in_sizes, int n_in,
                              void* d_out, int out_size, void* d_ws, size_t ws_size,
                              hipStream_t stream)
{
  const float* X  = (const float*)d_in[0];   // node_features [50000,128]
  const int*   EI = (const int*)d_in[1];     // edge_index    [2,800000]
  const float* Wt = (const float*)d_in[2];   // W             [2,128,128]
  const float* Bs = (const float*)d_in[3];   // b             [2,128]
  const float* Gm = (const float*)d_in[4];   // gamma         [2,128]
  const float* Bt = (const float*)d_in[5];   // beta          [2,128]
  float* out = (float*)d_out;

  const int n_nodes = in_sizes[0] / HIDDEN;
  const int n_edges = in_sizes[1] / 2;
  const int* src = EI;
  const int* dst = EI + n_edges;

  float*    hx  = (float*)d_ws;
  unsigned* agg = (unsigned*)((char*)d_ws + (size_t)n_nodes * HIDDEN * sizeof(float));

  const dim3 blk(256);
  const dim3 gemm_grid((((n_nodes + 15) / 16) + 7) / 8);     // 8 waves/block, 1 wave = 16 rows
  const dim3 elem_grid(((n_nodes * (HIDDEN / 4)) + 255) / 256);
  const dim3 edge_grid(4096);                                 // grid-stride, 32768 waves
  const dim3 node_grid((n_nodes + 7) / 8);                    // 1 wave = 1 node row

  const float* h = X;
  for (int layer = 0; layer < 2; ++layer) {
    gemm_bias_wmma<<<gemm_grid, blk, 0, stream>>>(
        h, Wt + (size_t)layer * HIDDEN * HIDDEN, Bs + layer * HIDDEN, hx, n_nodes);
    init_agg<<<elem_grid, blk, 0, stream>>>(hx, agg, n_nodes * (HIDDEN / 4));
    scatter_max_edges<<<edge_grid, blk, 0, stream>>>(hx, src, dst, agg, n_edges);
    ln_residual<<<node_grid, blk, 0, stream>>>(
        agg, h, Gm + layer * HIDDEN, Bt + layer * HIDDEN, out, n_nodes,
        (layer == 0) ? 1 : 0);
    h = out;   // layer-1 residual aliases d_out: each wave reads its row before writing it
  }
}